// RecurrentGCN_classification_31937376813750
// MI455X (gfx1250) — compile-verified
//
#include <hip/hip_runtime.h>
#include <hip/hip_bf16.h>
#include <math.h>

// ---------------------------------------------------------------------------
// DCRNN (RecurrentGCN classification) for gfx1250.
// Exploits Hst==0: R-gate dconv is dead, XH == XRH == [x, 0], so only the
// first 12 rows of each [44,32] weight slice contribute.
// Pipeline:
//   1) weighted degree segment-sums (atomics) + inversion
//   2) 8 sparse Chebyshev propagations into Tall[N][108]
//      (col blocks: x, T1o, T1i, T2o, T2i, T3o, T3i, T4o, T4i)
//   3) fused WMMA kernel: [16,108]@[108,32] (f32 16x16x4 WMMA, 27 k-steps)
//      for both Z and H_tilde, GRU combine, ReLU, [32]->[4] linear, pooling
//   4) divide pooled sums by counts
// ---------------------------------------------------------------------------

typedef float v2f __attribute__((ext_vector_type(2)));
typedef float v8f __attribute__((ext_vector_type(8)));

#define NN     100000
#define EE     1600000
#define FIN    12
#define FOUT   32
#define KTOT   108           // 9 term-blocks * 12 features
#define NCLS   4
#define NGRAPH 100
#define NTILES (NN / 16)     // 6250

// ---------------- init: Tall[:,0:12]=x, Tall[:,12:36]=0, deg=0 -------------
__global__ void init_kernel(const float* __restrict__ x, float* __restrict__ Tall,
                            float* __restrict__ degO, float* __restrict__ degI) {
  int i = blockIdx.x * blockDim.x + threadIdx.x;
  if (i >= NN) return;
  float* tr = Tall + (size_t)i * KTOT;
#pragma unroll
  for (int f = 0; f < 12; ++f) tr[f] = x[i * 12 + f];
#pragma unroll
  for (int f = 12; f < 36; ++f) tr[f] = 0.0f;
  degO[i] = 0.0f;
  degI[i] = 0.0f;
}

// ---------------- weight packing [2][5][44][32] -> [108][32] + zero pools --
__device__ __forceinline__ int widx(int dir, int k, int f, int c) {
  return ((dir * 5 + k) * 44 + f) * 32 + c;
}
__global__ void pack_zero_kernel(const float* __restrict__ Wz_in,
                                 const float* __restrict__ Wh_in,
                                 float* __restrict__ WzP, float* __restrict__ WhP,
                                 float* __restrict__ pool, float* __restrict__ cnt) {
  int i = blockIdx.x * blockDim.x + threadIdx.x;
  if (i < NGRAPH * NCLS) pool[i] = 0.0f;
  if (i < NGRAPH) cnt[i] = 0.0f;
  if (i >= KTOT * 32) return;
  int kr = i >> 5, c = i & 31;
  int t = kr / 12, f = kr % 12;
  float vz, vh;
  if (t == 0) {                       // k=0: W[0,0] + W[1,0]
    vz = Wz_in[widx(0, 0, f, c)] + Wz_in[widx(1, 0, f, c)];
    vh = Wh_in[widx(0, 0, f, c)] + Wh_in[widx(1, 0, f, c)];
  } else {                            // odd t -> fwd(dir0), even t -> rev(dir1)
    int dir = (t & 1) ? 0 : 1;
    int k = (t + 1) / 2;
    vz = Wz_in[widx(dir, k, f, c)];
    vh = Wh_in[widx(dir, k, f, c)];
  }
  WzP[i] = vz;
  WhP[i] = vh;
}

// ---------------- weighted degree accumulation -----------------------------
__global__ void deg_kernel(const int* __restrict__ row, const int* __restrict__ col,
                           const float* __restrict__ w,
                           float* __restrict__ degO, float* __restrict__ degI) {
  int e = blockIdx.x * blockDim.x + threadIdx.x;
  if (e >= EE) return;
  float we = w[e];
  atomicAdd(&degO[row[e]], we);
  atomicAdd(&degI[col[e]], we);
}

__global__ void inv_kernel(float* __restrict__ degO, float* __restrict__ degI) {
  int i = blockIdx.x * blockDim.x + threadIdx.x;
  if (i >= NN) return;
  degO[i] = 1.0f / degO[i];
  degI[i] = 1.0f / degI[i];
}

// ---------------- Chebyshev seed: T_{k} block <- -T_{k-2} block ------------
__global__ void seed_kernel(float* __restrict__ Tall, int outO, int srcO,
                            int outI, int srcI) {
  int i = blockIdx.x * blockDim.x + threadIdx.x;
  if (i >= NN) return;
  float* tr = Tall + (size_t)i * KTOT;
#pragma unroll
  for (int f = 0; f < 12; ++f) {
    tr[outO + f] = -tr[srcO + f];
    tr[outI + f] = -tr[srcI + f];
  }
}

// ---------------- fused fwd + rev propagation over edges -------------------
// fwd: out[col[e]][outO] += scale * degOinv[row[e]] * Tall[row[e]][inO]
// rev: out[row[e]][outI] += scale * degIinv[row[e]] * Tall[col[e]][inI]
__global__ void prop_kernel(float* __restrict__ Tall,
                            const int* __restrict__ row, const int* __restrict__ col,
                            const float* __restrict__ degOinv,
                            const float* __restrict__ degIinv,
                            int inO, int outO, int inI, int outI, float scale) {
  int e = blockIdx.x * blockDim.x + threadIdx.x;
  if (e >= EE) return;
  int s = row[e], d = col[e];
  {
    float nf = scale * degOinv[s];
    const float4* hs = (const float4*)(Tall + (size_t)s * KTOT + inO);
    float* od = Tall + (size_t)d * KTOT + outO;
    float4 h0 = hs[0], h1 = hs[1], h2 = hs[2];
    atomicAdd(od + 0, nf * h0.x);  atomicAdd(od + 1, nf * h0.y);
    atomicAdd(od + 2, nf * h0.z);  atomicAdd(od + 3, nf * h0.w);
    atomicAdd(od + 4, nf * h1.x);  atomicAdd(od + 5, nf * h1.y);
    atomicAdd(od + 6, nf * h1.z);  atomicAdd(od + 7, nf * h1.w);
    atomicAdd(od + 8, nf * h2.x);  atomicAdd(od + 9, nf * h2.y);
    atomicAdd(od + 10, nf * h2.z); atomicAdd(od + 11, nf * h2.w);
  }
  {
    float nr = scale * degIinv[s];
    const float4* hd = (const float4*)(Tall + (size_t)d * KTOT + inI);
    float* os = Tall + (size_t)s * KTOT + outI;
    float4 h0 = hd[0], h1 = hd[1], h2 = hd[2];
    atomicAdd(os + 0, nr * h0.x);  atomicAdd(os + 1, nr * h0.y);
    atomicAdd(os + 2, nr * h0.z);  atomicAdd(os + 3, nr * h0.w);
    atomicAdd(os + 4, nr * h1.x);  atomicAdd(os + 5, nr * h1.y);
    atomicAdd(os + 6, nr * h1.z);  atomicAdd(os + 7, nr * h1.w);
    atomicAdd(os + 8, nr * h2.x);  atomicAdd(os + 9, nr * h2.y);
    atomicAdd(os + 10, nr * h2.z); atomicAdd(os + 11, nr * h2.w);
  }
}

// ---------------- fused WMMA GEMM + GRU + linear + pool --------------------
// One wave per 16-node tile. A = Tall[16,108] f32, B = packed weights [108,32].
// v_wmma_f32_16x16x4_f32, K=108 -> 27 steps, 4 accumulators (Z/H x 2 tiles).
__global__ void __launch_bounds__(256)
fused_gru_kernel(const float* __restrict__ Tall,
                 const float* __restrict__ WzP, const float* __restrict__ WhP,
                 const float* __restrict__ bz, const float* __restrict__ bh,
                 const float* __restrict__ Wlin, const float* __restrict__ blin,
                 const int* __restrict__ batch,
                 float* __restrict__ pool, float* __restrict__ cnt) {
  __shared__ float lds[8][16][32];
  int lane = threadIdx.x & 31;
  int wave = threadIdx.x >> 5;
  int tile = blockIdx.x * 8 + wave;
  bool active = (tile < NTILES);            // wave-uniform: EXEC stays all-1s
  int node0 = tile * 16;
  int m = lane & 15;                        // M (for A) / N (for B,C)
  int kh = lane >> 4;                       // K-half selector

  if (active) {
    v8f cz0 = {}, cz1 = {}, ch0 = {}, ch1 = {};
    const float* Arow = Tall + (size_t)(node0 + m) * KTOT;
    for (int kk = 0; kk < 27; ++kk) {
      int kb = 4 * kk + 2 * kh;
      v2f a;
      a.x = Arow[kb];
      a.y = Arow[kb + 1];
      v2f b0, b1;
      b0.x = WzP[kb * 32 + m];        b0.y = WzP[(kb + 1) * 32 + m];
      b1.x = WzP[kb * 32 + 16 + m];   b1.y = WzP[(kb + 1) * 32 + 16 + m];
      cz0 = __builtin_amdgcn_wmma_f32_16x16x4_f32(false, a, false, b0, (short)0, cz0, false, false);
      cz1 = __builtin_amdgcn_wmma_f32_16x16x4_f32(false, a, false, b1, (short)0, cz1, false, false);
      b0.x = WhP[kb * 32 + m];        b0.y = WhP[(kb + 1) * 32 + m];
      b1.x = WhP[kb * 32 + 16 + m];   b1.y = WhP[(kb + 1) * 32 + 16 + m];
      ch0 = __builtin_amdgcn_wmma_f32_16x16x4_f32(false, a, false, b0, (short)0, ch0, false, false);
      ch1 = __builtin_amdgcn_wmma_f32_16x16x4_f32(false, a, false, b1, (short)0, ch1, false, false);
    }
    // epilogue: Z = sigmoid(.+bz), Ht = tanh(.+bh), H = (1-Z)*Ht, ReLU
    float bzl0 = bz[m], bzl1 = bz[16 + m];
    float bhl0 = bh[m], bhl1 = bh[16 + m];
#pragma unroll
    for (int r = 0; r < 8; ++r) {
      int nl = kh ? (r + 8) : r;            // local node row
      float z0 = 1.0f / (1.0f + expf(-(cz0[r] + bzl0)));
      float z1 = 1.0f / (1.0f + expf(-(cz1[r] + bzl1)));
      float h0 = (1.0f - z0) * tanhf(ch0[r] + bhl0);
      float h1 = (1.0f - z1) * tanhf(ch1[r] + bhl1);
      lds[wave][nl][m]      = fmaxf(h0, 0.0f);
      lds[wave][nl][16 + m] = fmaxf(h1, 0.0f);
    }
  }
  __syncthreads();
  if (active) {
    // each lane: one node (lane>>1), two classes (2*(lane&1), +1)
    int r = lane >> 1;
    int c0 = 2 * (lane & 1);
    float a0 = blin[c0], a1 = blin[c0 + 1];
#pragma unroll
    for (int f = 0; f < 32; ++f) {
      float h = lds[wave][r][f];
      a0 += h * Wlin[f * 4 + c0];
      a1 += h * Wlin[f * 4 + c0 + 1];
    }
    int g = batch[node0 + r];
    atomicAdd(&pool[g * 4 + c0], a0);
    atomicAdd(&pool[g * 4 + c0 + 1], a1);
    if ((lane & 1) == 0) atomicAdd(&cnt[g], 1.0f);
  }
}

// ---------------- mean pool finalize ---------------------------------------
__global__ void final_kernel(const float* __restrict__ pool,
                             const float* __restrict__ cnt,
                             float* __restrict__ out) {
  int i = blockIdx.x * blockDim.x + threadIdx.x;
  if (i >= NGRAPH * NCLS) return;
  out[i] = pool[i] / cnt[i >> 2];
}

// ---------------------------------------------------------------------------
extern "C" void kernel_launch(void* const* d_in, const int* in_sizes, int n_in,
                              void* d_out, int out_size, void* d_ws, size_t ws_size,
                              hipStream_t stream) {
  const float* x     = (const float*)d_in[0];
  const int*   ei    = (const int*)d_in[1];
  const int*   row   = ei;
  const int*   col   = ei + EE;
  const float* ew    = (const float*)d_in[2];
  const int*   batch = (const int*)d_in[3];
  const float* W_z   = (const float*)d_in[4];
  const float* b_z   = (const float*)d_in[5];
  // d_in[6], d_in[7] = W_r, b_r: dead (R*Hst == 0)
  const float* W_h   = (const float*)d_in[8];
  const float* b_h   = (const float*)d_in[9];
  const float* W_lin = (const float*)d_in[10];
  const float* b_lin = (const float*)d_in[11];
  float* out = (float*)d_out;

  float* ws   = (float*)d_ws;
  float* Tall = ws;                              // N * 108
  float* degO = Tall + (size_t)NN * KTOT;        // N
  float* degI = degO + NN;                       // N
  float* WzP  = degI + NN;                       // 108*32
  float* WhP  = WzP + KTOT * 32;                 // 108*32
  float* pool = WhP + KTOT * 32;                 // G*C
  float* cntb = pool + NGRAPH * NCLS;            // G

  const int T = 256;
  dim3 b(T);
  dim3 gN((NN + T - 1) / T);
  dim3 gE((EE + T - 1) / T);

  init_kernel<<<gN, b, 0, stream>>>(x, Tall, degO, degI);
  pack_zero_kernel<<<(KTOT * 32 + T - 1) / T, b, 0, stream>>>(W_z, W_h, WzP, WhP, pool, cntb);
  deg_kernel<<<gE, b, 0, stream>>>(row, col, ew, degO, degI);
  inv_kernel<<<gN, b, 0, stream>>>(degO, degI);

  // T1o@12 = prop_fwd(x@0), T1i@24 = prop_rev(x@0)
  prop_kernel<<<gE, b, 0, stream>>>(Tall, row, col, degO, degI, 0, 12, 0, 24, 1.0f);
  // T2 = 2*prop(T1) - T0   (T0 = x)
  seed_kernel<<<gN, b, 0, stream>>>(Tall, 36, 0, 48, 0);
  prop_kernel<<<gE, b, 0, stream>>>(Tall, row, col, degO, degI, 12, 36, 24, 48, 2.0f);
  // T3 = 2*prop(T2) - T1
  seed_kernel<<<gN, b, 0, stream>>>(Tall, 60, 12, 72, 24);
  prop_kernel<<<gE, b, 0, stream>>>(Tall, row, col, degO, degI, 36, 60, 48, 72, 2.0f);
  // T4 = 2*prop(T3) - T2
  seed_kernel<<<gN, b, 0, stream>>>(Tall, 84, 36, 96, 48);
  prop_kernel<<<gE, b, 0, stream>>>(Tall, row, col, degO, degI, 60, 84, 72, 96, 2.0f);

  // fused WMMA GEMMs + GRU + final linear + pooling
  fused_gru_kernel<<<(NTILES + 7) / 8, b, 0, stream>>>(
      Tall, WzP, WhP, b_z, b_h, W_lin, b_lin, batch, pool, cntb);

  final_kernel<<<(NGRAPH * NCLS + T - 1) / T, b, 0, stream>>>(pool, cntb, out);
}